// GetScalars_19035295055889
// MI455X (gfx1250) — compile-verified
//
#include <hip/hip_runtime.h>

// ---------------------------------------------------------------------------
// GetScalars for MI455X (gfx1250): per-atom Gram matrices via f32 WMMA,
// LDS transpose for coalesced output stores along the atom-fast axis.
//
// A = 20000 atoms, C = 32 channels, l = 0..3 (K = 1,3,5,7 padded to 4/8).
// Output: rows [0,32)              = raw copy of parts0 (reshape is a no-op)
//         rows [32+l*1024, +1024)  = G_l[c][d] at row 32 + l*1024 + c*32 + d
// ---------------------------------------------------------------------------

typedef __attribute__((ext_vector_type(2))) float v2f;
typedef __attribute__((ext_vector_type(8))) float v8f;

#define A_NUM        20000
#define C_NUM        32
#define ATOMS_PER_WG 32          // one wave per atom, 32 waves = 1024 threads
#define ATOM_STRIDE  257         // dwords; odd => conflict-free transposed reads
#define NTHREADS     1024

__global__ __launch_bounds__(NTHREADS)
void gram_wmma_kernel(const float* __restrict__ p0,
                      const float* __restrict__ p1,
                      const float* __restrict__ p2,
                      const float* __restrict__ p3,
                      float* __restrict__ out)
{
    __shared__ float lds[ATOMS_PER_WG * ATOM_STRIDE];   // ~32.9 KB

    const int tid   = threadIdx.x;
    const int wave  = tid >> 5;        // 0..31 : local atom in compute phase
    const int lane  = tid & 31;
    const int hi    = lane >> 4;       // half-wave selector (K rows 2,3)
    const int l15   = lane & 15;

    const int atom0 = blockIdx.x * ATOMS_PER_WG;
    const int atom  = atom0 + wave;

    // ---- block 0: raw flat copy of parts0 (reshape preserves memory order)
    {
        const int idx = atom0 * C_NUM + tid;   // 32 atoms * 32 ch == 1024
        out[idx] = p0[idx];
    }

    const float* const parts[4] = { p0, p1, p2, p3 };

    #pragma unroll
    for (int l = 0; l < 4; ++l) {
        const int M = 2 * l + 1;               // K extent (1,3,5,7)
        const int G = (M + 3) / 4;             // K-groups of 4 (1,1,2,2)
        const float* pb = parts[l] + (size_t)atom * M * C_NUM;

        __builtin_prefetch(pb, 0, 0);          // global_prefetch_b8

        // Operand op[t][g]: serves as BOTH A (row tile t) and B (col tile t).
        // lane<16 : v[0]=p[4g+0][16t+l15], v[1]=p[4g+1][16t+l15]
        // lane>=16: v[0]=p[4g+2][16t+l15], v[1]=p[4g+3][16t+l15]
        v2f op[2][2];
        #pragma unroll
        for (int t = 0; t < 2; ++t) {
            #pragma unroll
            for (int g = 0; g < 2; ++g) {
                float e0 = 0.0f, e1 = 0.0f;
                if (g < G) {
                    const int m0  = 4 * g + 2 * hi;
                    const int col = 16 * t + l15;
                    if (m0     < M) e0 = pb[(size_t)(m0    ) * C_NUM + col];
                    if (m0 + 1 < M) e1 = pb[(size_t)(m0 + 1) * C_NUM + col];
                }
                op[t][g][0] = e0;
                op[t][g][1] = e1;
            }
        }

        const size_t rowBase = (size_t)(C_NUM + l * C_NUM * C_NUM);

        #pragma unroll
        for (int ct = 0; ct < 2; ++ct) {
            #pragma unroll
            for (int dt = 0; dt < 2; ++dt) {
                // D = op[ct]^ * op[dt]  (Gram tile), K accumulated in groups of 4
                v8f acc = {};
                acc = __builtin_amdgcn_wmma_f32_16x16x4_f32(
                          false, op[ct][0], false, op[dt][0],
                          (short)0, acc, false, false);
                if (G == 2)
                    acc = __builtin_amdgcn_wmma_f32_16x16x4_f32(
                              false, op[ct][1], false, op[dt][1],
                              (short)0, acc, false, false);

                // ---- stage tile to LDS as [atom][row_local], row_local = M*16+N
                __syncthreads();                     // prior reads complete
                #pragma unroll
                for (int r = 0; r < 8; ++r) {
                    const int row_local = (r + 8 * hi) * 16 + l15; // (M,N)
                    lds[wave * ATOM_STRIDE + row_local] = acc[r];
                }
                __syncthreads();

                // ---- coalesced global stores: lane spans atoms (128B/wave-store)
                #pragma unroll
                for (int i = 0; i < 8; ++i) {
                    const int row_local = wave * 8 + i;      // 32 waves * 8 = 256
                    const int Mr = row_local >> 4;
                    const int Nc = row_local & 15;
                    const size_t grow = rowBase
                                      + (size_t)(ct * 16 + Mr) * C_NUM
                                      + (size_t)(dt * 16 + Nc);
                    out[grow * (size_t)A_NUM + (size_t)(atom0 + lane)] =
                        lds[lane * ATOM_STRIDE + row_local];
                }
            }
        }
    }
}

extern "C" void kernel_launch(void* const* d_in, const int* in_sizes, int n_in,
                              void* d_out, int out_size, void* d_ws, size_t ws_size,
                              hipStream_t stream) {
    (void)in_sizes; (void)n_in; (void)out_size; (void)d_ws; (void)ws_size;
    const float* p0 = (const float*)d_in[0];
    const float* p1 = (const float*)d_in[1];
    const float* p2 = (const float*)d_in[2];
    const float* p3 = (const float*)d_in[3];
    float* out = (float*)d_out;

    dim3 grid(A_NUM / ATOMS_PER_WG);   // 625 workgroups
    dim3 block(NTHREADS);              // 32 waves (wave32)
    gram_wmma_kernel<<<grid, block, 0, stream>>>(p0, p1, p2, p3, out);
}